// Grad_80101140070842
// MI455X (gfx1250) — compile-verified
//
#include <hip/hip_runtime.h>
#include <stdint.h>

#define THREADS 256
#define NEPS 1e-12f

// Low 32 bits of a generic pointer into LDS == LDS byte offset (LDS aperture
// truncation rule, ISA 10.2) — exactly what the async-store VSRC wants.
__device__ __forceinline__ unsigned lds_off(const void* p) {
  return (unsigned)(uintptr_t)p;
}

__global__ __launch_bounds__(THREADS) void grad_edge_kernel(
    const float* __restrict__ cell,   // [nbatch*9]
    const int*   __restrict__ bt,     // [T]
    const float* __restrict__ eij,    // [T*3]
    const float* __restrict__ eik,    // [T*3]
    float*       __restrict__ out,    // [T*27]
    int T, int cellFloats)
{
  __shared__ __align__(16) float sCell[576];          // 64 * 3*3 = 2304 B
  __shared__ __align__(16) float stage[THREADS * 27]; // 27,648 B, wave-sliced

  const int tid   = threadIdx.x;
  const int nCell = cellFloats < 576 ? cellFloats : 576;

  // Cooperative cell preload, loop-free: nV4 <= 144 < THREADS, tail <= 3.
  const int nV4 = nCell >> 2;
  if (tid < nV4)
    ((float4*)sCell)[tid] = ((const float4*)cell)[tid];
  if (tid < (nCell & 3))
    sCell[(nV4 << 2) + tid] = cell[(nV4 << 2) + tid];
  __syncthreads();

  const int lane      = tid & 31;
  const int wave      = tid >> 5;
  const int waveFirst = blockIdx.x * THREADS + wave * 32;
  if (waveFirst >= T) return;                 // fully inactive wave

  const int t  = waveFirst + lane;
  const int tc = (t < T) ? t : (T - 1);       // clamp for (rare) partial wave

  const int b = bt[tc];
  const float* R = &sCell[b * 9];
  const float r00=R[0], r01=R[1], r02=R[2];
  const float r10=R[3], r11=R[4], r12=R[5];
  const float r20=R[6], r21=R[7], r22=R[8];

  const float ex = eij[3*tc+0], ey = eij[3*tc+1], ez = eij[3*tc+2];
  const float fx = eik[3*tc+0], fy = eik[3*tc+1], fz = eik[3*tc+2];

  // u = rho @ e_ij ; v = rho @ e_ik
  const float u0 = r00*ex + r01*ey + r02*ez;
  const float u1 = r10*ex + r11*ey + r12*ez;
  const float u2 = r20*ex + r21*ey + r22*ez;
  const float v0 = r00*fx + r01*fy + r02*fz;
  const float v1 = r10*fx + r11*fy + r12*fz;
  const float v2 = r20*fx + r21*fy + r22*fz;

  float* st = &stage[tid * 27];   // stride 27 dwords: coprime with 64 banks

  // ---- grad_distance(u): outer(normalize(u), u)
  {
    const float n  = sqrtf(u0*u0 + u1*u1 + u2*u2);
    const float in = 1.0f / fmaxf(n, NEPS);
    const float a0 = u0*in, a1 = u1*in, a2 = u2*in;
    st[0]=a0*u0; st[1]=a0*u1; st[2]=a0*u2;
    st[3]=a1*u0; st[4]=a1*u1; st[5]=a1*u2;
    st[6]=a2*u0; st[7]=a2*u1; st[8]=a2*u2;
  }
  // ---- grad_distance(v): outer(normalize(v), v)
  {
    const float n  = sqrtf(v0*v0 + v1*v1 + v2*v2);
    const float in = 1.0f / fmaxf(n, NEPS);
    const float b0 = v0*in, b1 = v1*in, b2 = v2*in;
    st[9] =b0*v0; st[10]=b0*v1; st[11]=b0*v2;
    st[12]=b1*v0; st[13]=b1*v1; st[14]=b1*v2;
    st[15]=b2*v0; st[16]=b2*v1; st[17]=b2*v2;
  }
  // ---- grad_angle
  {
    const float c0 = u1*v2 - u2*v1;
    const float c1 = u2*v0 - u0*v2;
    const float c2 = u0*v1 - u1*v0;
    const float y  = sqrtf(c0*c0 + c1*c1 + c2*c2);
    const float xd = u0*v0 + u1*v1 + u2*v2;
    const float id = 1.0f / (xd*xd + y*y);
    const float ay = xd * id;
    const float ax = -y * id;
    const float ic = 1.0f / fmaxf(y, NEPS);
    const float d0 = c0*ic, d1 = c1*ic, d2 = c2*ic;      // dn
    // dcn_u = cross(v, dn) ; dcn_v = cross(dn, u)   (einsum with Levi-Civita K)
    const float pu0 = v1*d2 - v2*d1;
    const float pu1 = v2*d0 - v0*d2;
    const float pu2 = v0*d1 - v1*d0;
    const float pv0 = d1*u2 - d2*u1;
    const float pv1 = d2*u0 - d0*u2;
    const float pv2 = d0*u1 - d1*u0;
    const float du0 = ay*pu0 + ax*v0;
    const float du1 = ay*pu1 + ax*v1;
    const float du2 = ay*pu2 + ax*v2;
    const float dv0 = ay*pv0 + ax*u0;
    const float dv1 = ay*pv1 + ax*u1;
    const float dv2 = ay*pv2 + ax*u2;
    st[18]=du0*u0+dv0*v0; st[19]=du0*u1+dv0*v1; st[20]=du0*u2+dv0*v2;
    st[21]=du1*u0+dv1*v0; st[22]=du1*u1+dv1*v1; st[23]=du1*u2+dv1*v2;
    st[24]=du2*u0+dv2*v0; st[25]=du2*u1+dv2*v1; st[26]=du2*u2+dv2*v2;
  }

  if (waveFirst + 32 <= T) {
    // Full wave: drain the 3456-B (216 x 16B) staged tile straight from LDS to
    // HBM with async b128 stores (ASYNCcnt path, no VGPR round-trip).
    // Output is write-once/never-read and bigger than L2 -> non-temporal.
    asm volatile("s_wait_dscnt 0x0" ::: "memory");     // LDS writes landed
    const unsigned ldsBase = lds_off(&stage[wave * 32 * 27]);
    const unsigned gBase   = (unsigned)waveFirst * 108u; // byte offset, < 2^31
#pragma unroll
    for (int k = 0; k < 6; ++k) {
      const unsigned off = (unsigned)(k * 32 + lane) * 16u;
      asm volatile("global_store_async_from_lds_b128 %0, %1, %2 th:TH_STORE_NT"
                   :: "v"(gBase + off), "v"(ldsBase + off), "s"(out)
                   : "memory");
    }
    if (lane < 24) {                                    // 216 = 6*32 + 24
      const unsigned off = (unsigned)(192 + lane) * 16u;
      asm volatile("global_store_async_from_lds_b128 %0, %1, %2 th:TH_STORE_NT"
                   :: "v"(gBase + off), "v"(ldsBase + off), "s"(out)
                   : "memory");
    }
    // s_endpgm's implicit wait-idle drains ASYNCcnt before LDS dealloc.
  } else if (t < T) {
    // Partial-wave fallback (not hit for T = 2,000,000)
#pragma unroll
    for (int m = 0; m < 27; ++m) out[(size_t)t * 27 + m] = st[m];
  }
}

extern "C" void kernel_launch(void* const* d_in, const int* in_sizes, int n_in,
                              void* d_out, int out_size, void* d_ws, size_t ws_size,
                              hipStream_t stream) {
  const float* cell = (const float*)d_in[0];   // [64,3,3] f32
  const int*   bt   = (const int*)d_in[1];     // [T] i32
  const float* eij  = (const float*)d_in[2];   // [T,3] f32
  const float* eik  = (const float*)d_in[3];   // [T,3] f32
  float*       out  = (float*)d_out;           // [T,3,3,3] f32

  const int cellFloats = in_sizes[0];
  const int T          = in_sizes[1];
  const int blocks     = (T + THREADS - 1) / THREADS;
  grad_edge_kernel<<<blocks, THREADS, 0, stream>>>(cell, bt, eij, eik, out,
                                                   T, cellFloats);
}